// ResidualVectorQuantizerEMA_87883620811553
// MI455X (gfx1250) — compile-verified
//
#include <hip/hip_runtime.h>
#include <hip/hip_bf16.h>

typedef __attribute__((ext_vector_type(2))) float v2f;
typedef __attribute__((ext_vector_type(4))) float v4f;
typedef __attribute__((ext_vector_type(8))) float v8f;

#define N_TOKENS 131072
#define E_DIM    128
#define CB_SIZE  1024
#define NUM_Q    4
#define BETA     0.25f

// LDS codebook chunk: 64 codewords x 128 floats, padded row stride 132 floats
// -> bank(c*132+k) = (4c+k) & 63 : the 32-lane float2 B-fragment read pattern
//    covers all 64 banks exactly once (conflict-free). Double-buffered.
#define CHUNK_CW   64
#define LDS_STRIDE 132
#define NCHUNK     (CB_SIZE / CHUNK_CW)

// ---------------------------------------------------------------------------
// Precompute ||c||^2 for all 4x1024 codewords into ws[0..4095]
// ---------------------------------------------------------------------------
__global__ __launch_bounds__(256) void rvq_cnorm_kernel(
    const float* __restrict__ cb, float* __restrict__ ws) {
  int i = blockIdx.x * 256 + threadIdx.x;  // 0..4095
  const float* row = cb + (size_t)i * E_DIM;
  float s = 0.f;
  #pragma unroll
  for (int k = 0; k < E_DIM; k += 4) {
    v4f v = *(const v4f*)(row + k);
    s += v.x * v.x + v.y * v.y + v.z * v.z + v.w * v.w;
  }
  ws[i] = s;
}

// ---------------------------------------------------------------------------
// Fused 4-stage RVQ. One block = 128 tokens (8 waves x 16 rows).
// Residual + x_q live in registers across all stages (x read once from HBM,
// x_q written once -> ~130MB total HBM traffic, the roofline floor).
// Distance GEMM on v_wmma_f32_16x16x4_f32; codebook double-buffered in LDS.
// ---------------------------------------------------------------------------
__global__ __launch_bounds__(256) void rvq_main_kernel(
    const float* __restrict__ x, const float* __restrict__ cb,
    float* __restrict__ out, float* __restrict__ ws) {
  __shared__ float lds_cb[2][CHUNK_CW * LDS_STRIDE];
  __shared__ int   lds_idx[128];
  __shared__ float lds_loss[8];

  const int t    = threadIdx.x;
  const int w    = t >> 5;        // wave id 0..7
  const int lane = t & 31;
  const int half = lane >> 4;     // 0: K%4 in {0,1}; 1: K%4 in {2,3}
  const int lm   = lane & 15;
  const int rowbase = blockIdx.x * 128 + w * 16;
  const int myrow   = rowbase + lm;         // A-fragment row for this lane

  // ---- load x into residual A-fragments (ISA 16x4 f32 A layout) ----
  v2f r[32], xq[32];
  #pragma unroll
  for (int f = 0; f < 32; ++f) {
    const int k = 4 * f + 2 * half;
    r[f]  = *(const v2f*)(x + (size_t)myrow * E_DIM + k);
    xq[f] = v2f{0.f, 0.f};
  }
  float lossacc = 0.f;

  for (int s = 0; s < NUM_Q; ++s) {
    const float* cbs = cb + (size_t)s * CB_SIZE * E_DIM;

    float bestd[8];
    int   besti[8];
    #pragma unroll
    for (int v = 0; v < 8; ++v) { bestd[v] = 3.402823e38f; besti[v] = 0; }

    // prologue: stage chunk 0 into buffer 0
    {
      const float* src = cbs;
      #pragma unroll
      for (int u = 0; u < 8; ++u) {
        const int i4  = t * 8 + u;
        const int row = i4 >> 5;
        const int col = (i4 * 4) & 127;
        *(v4f*)(&lds_cb[0][row * LDS_STRIDE + col]) = *(const v4f*)(src + i4 * 4);
      }
    }
    __syncthreads();

    for (int ch = 0; ch < NCHUNK; ++ch) {
      const int cur = ch & 1;

      // issue global loads for NEXT chunk now; their latency hides behind
      // the 128-wmma compute below (ds_store + wait happen after compute)
      v4f nxt[8];
      if (ch + 1 < NCHUNK) {
        const float* src = cbs + (size_t)(ch + 1) * CHUNK_CW * E_DIM;
        #pragma unroll
        for (int u = 0; u < 8; ++u) nxt[u] = *(const v4f*)(src + (t * 8 + u) * 4);
      }

      #pragma unroll
      for (int sub = 0; sub < 4; ++sub) {
        // preload the whole B panel into distinct registers first so the
        // WMMA chain isn't serialized on per-load s_wait_dscnt
        v2f bfr[32];
        const float* brow = &lds_cb[cur][(sub * 16 + lm) * LDS_STRIDE + 2 * half];
        #pragma unroll
        for (int f = 0; f < 32; ++f) bfr[f] = *(const v2f*)(brow + 4 * f);

        v8f acc0 = {0.f, 0.f, 0.f, 0.f, 0.f, 0.f, 0.f, 0.f};
        v8f acc1 = {0.f, 0.f, 0.f, 0.f, 0.f, 0.f, 0.f, 0.f};
        #pragma unroll
        for (int f = 0; f < 32; f += 2) {
          acc0 = __builtin_amdgcn_wmma_f32_16x16x4_f32(
              false, r[f],     false, bfr[f],     (short)0, acc0, false, false);
          acc1 = __builtin_amdgcn_wmma_f32_16x16x4_f32(
              false, r[f + 1], false, bfr[f + 1], (short)0, acc1, false, false);
        }
        const int cw0 = ch * CHUNK_CW + sub * 16;
        const float cn = ws[s * CB_SIZE + cw0 + lm];  // ||c||^2 for col lm
        #pragma unroll
        for (int v = 0; v < 8; ++v) {
          float d = cn - 2.0f * (acc0[v] + acc1[v]);
          if (d < bestd[v]) { bestd[v] = d; besti[v] = cw0 + lm; }
        }
      }

      // store the prefetched chunk into the other buffer, then one barrier
      if (ch + 1 < NCHUNK) {
        #pragma unroll
        for (int u = 0; u < 8; ++u) {
          const int i4  = t * 8 + u;
          const int row = i4 >> 5;
          const int col = (i4 * 4) & 127;
          *(v4f*)(&lds_cb[1 - cur][row * LDS_STRIDE + col]) = nxt[u];
        }
      }
      __syncthreads();
    }

    // argmin across the 16-lane column groups (ties -> lowest index,
    // matching jnp.argmin first-occurrence semantics)
    #pragma unroll
    for (int v = 0; v < 8; ++v) {
      #pragma unroll
      for (int m = 1; m < 16; m <<= 1) {
        float od = __shfl_xor(bestd[v], m, 32);
        int   oi = __shfl_xor(besti[v], m, 32);
        if (od < bestd[v] || (od == bestd[v] && oi < besti[v])) {
          bestd[v] = od; besti[v] = oi;
        }
      }
    }

    // lanes 0 / 16 publish per-row indices (rows v+8*half); same-wave LDS
    // ops are in-order, so the read below sees them without a barrier.
    if (lm == 0) {
      #pragma unroll
      for (int v = 0; v < 8; ++v) {
        lds_idx[w * 16 + half * 8 + v] = besti[v];
        const int rr = rowbase + half * 8 + v;
        out[(size_t)N_TOKENS * E_DIM + 1 + (size_t)rr * NUM_Q + s] =
            (float)besti[v];
      }
    }
    const int idxm = lds_idx[w * 16 + lm];

    // gather chosen codeword, update residual / x_q / loss
    const float* qrow = cbs + (size_t)idxm * E_DIM;
    #pragma unroll
    for (int f = 0; f < 32; ++f) {
      const int k = 4 * f + 2 * half;
      v2f q = *(const v2f*)(qrow + k);
      float d0 = r[f].x - q.x, d1 = r[f].y - q.y;
      lossacc += d0 * d0 + d1 * d1;
      r[f]  = v2f{d0, d1};
      xq[f].x += q.x; xq[f].y += q.y;
    }
  }

  // store x_q
  #pragma unroll
  for (int f = 0; f < 32; ++f) {
    const int k = 4 * f + 2 * half;
    *(v2f*)(out + (size_t)myrow * E_DIM + k) = xq[f];
  }

  // deterministic block loss partial -> ws[4096 + block]
  #pragma unroll
  for (int m = 1; m < 32; m <<= 1) lossacc += __shfl_xor(lossacc, m, 32);
  if (lane == 0) lds_loss[w] = lossacc;
  __syncthreads();
  if (t == 0) {
    float tot = 0.f;
    #pragma unroll
    for (int i = 0; i < 8; ++i) tot += lds_loss[i];
    ws[4096 + blockIdx.x] = tot;
  }
}

// ---------------------------------------------------------------------------
// Deterministic serial reduction of 1024 block partials -> mean loss
// ---------------------------------------------------------------------------
__global__ void rvq_finalize_kernel(const float* __restrict__ ws,
                                    float* __restrict__ out) {
  if (threadIdx.x == 0 && blockIdx.x == 0) {
    float tot = 0.f;
    for (int i = 0; i < 1024; ++i) tot += ws[4096 + i];
    // mean over stages of beta*mean(sq) = beta * total / (NUM_Q * N * E)
    out[(size_t)N_TOKENS * E_DIM] =
        BETA * tot / ((float)NUM_Q * (float)N_TOKENS * (float)E_DIM);
  }
}

extern "C" void kernel_launch(void* const* d_in, const int* in_sizes, int n_in,
                              void* d_out, int out_size, void* d_ws, size_t ws_size,
                              hipStream_t stream) {
  const float* x  = (const float*)d_in[0];   // [131072, 128] f32
  const float* cb = (const float*)d_in[1];   // [4, 1024, 128] f32
  float* out = (float*)d_out;                // xq | loss | indices(as f32)
  float* ws  = (float*)d_ws;                 // [0..4095] cnorm, [4096..5119] loss partials

  rvq_cnorm_kernel<<<(NUM_Q * CB_SIZE) / 256, 256, 0, stream>>>(cb, ws);
  rvq_main_kernel<<<N_TOKENS / 128, 256, 0, stream>>>(x, cb, out, ws);
  rvq_finalize_kernel<<<1, 64, 0, stream>>>(ws, out);
}